// NeuralOUEstimator_80796924772833
// MI455X (gfx1250) — compile-verified
//
#include <hip/hip_runtime.h>
#include <hip/hip_bf16.h>
#include <cstdint>

typedef float v2f __attribute__((ext_vector_type(2)));
typedef float v8f __attribute__((ext_vector_type(8)));

#define SEQ_T 2048
#define HID   8

__device__ __forceinline__ float fast_tanhf(float x) {
#if __has_builtin(__builtin_amdgcn_tanhf)
    return __builtin_amdgcn_tanhf(x);
#else
    float r;
    asm volatile("v_tanh_f32 %0, %1\n\tv_nop" : "=v"(r) : "v"(x));
    return r;
#endif
}

// lane L <-> lane L^16 exchange: ds_swizzle group-of-32, and=0x1f, or=0, xor=0x10
__device__ __forceinline__ float swapx16(float x) {
    return __int_as_float(__builtin_amdgcn_ds_swizzle(__float_as_int(x), 0x401F));
}

__device__ __forceinline__ float softplusf(float x) {
    return (x > 20.0f) ? x : logf(1.0f + __expf(x));
}

__global__ __launch_bounds__(64)
void NeuralOUEstimator_lstm_wmma_kernel(
        const float* __restrict__ seq,   // [B, T, 1]
        const float* __restrict__ W_ih,  // [32, 1]
        const float* __restrict__ W_hh,  // [32, 8]
        const float* __restrict__ b_ih,  // [32]
        const float* __restrict__ b_hh,  // [32]
        const float* __restrict__ W1p,   // [8, 8]
        const float* __restrict__ b1p,   // [8]
        const float* __restrict__ W2p,   // [3, 8]
        const float* __restrict__ b2p,   // [3]
        float* __restrict__ out,         // [B, 3]
        int Btotal)
{
    const int tid  = threadIdx.x;
    const int lane = tid & 31;
    const int l16  = lane & 15;
    const int hi   = lane >> 4;          // 0 = low half, 1 = high half
    const bool low = (hi == 0);

    const int waveG = blockIdx.x * (blockDim.x >> 5) + (tid >> 5);
    int s = waveG * 16 + l16;            // this lane's sample
    if (s >= Btotal) s = Btotal - 1;     // clamp keeps EXEC all-ones for WMMA

    // ---- A-matrix (constant): W_hh tiles in WMMA 16x4 f32 A-layout ----
    // A[tile t][chunk c], vgpr v, lane L -> W_hh[(16t + L%16)*8 + (4c + 2*hi + v)]
    v2f a00, a01, a10, a11;
    {
        const int r0 = l16 * HID;         // tile0: gates 0..15 (i,f)
        const int r1 = (16 + l16) * HID;  // tile1: gates 16..31 (g,o)
        const int k0 = 2 * hi;
        a00[0] = W_hh[r0 + k0];     a00[1] = W_hh[r0 + k0 + 1];
        a01[0] = W_hh[r0 + 4 + k0]; a01[1] = W_hh[r0 + 4 + k0 + 1];
        a10[0] = W_hh[r1 + k0];     a10[1] = W_hh[r1 + k0 + 1];
        a11[0] = W_hh[r1 + 4 + k0]; a11[1] = W_hh[r1 + 4 + k0 + 1];
    }

    // ---- input-gate constants in WMMA D-layout (vgpr k, this lane-half) ----
    float xw0[8], xb0[8], xw1[8], xb1[8];
    #pragma unroll
    for (int k = 0; k < 8; ++k) {
        const int g0 = k + 8 * hi;       // tile0 gate row
        const int g1 = 16 + g0;          // tile1 gate row
        xw0[k] = W_ih[g0]; xb0[k] = b_ih[g0] + b_hh[g0];
        xw1[k] = W_ih[g1]; xb1[k] = b_ih[g1] + b_hh[g1];
    }

    // tile1 activation: low lanes tanh(g), high lanes sigmoid(o)=0.5*tanh(0.5x)+0.5
    const float sclA = low ? 1.0f : 0.5f;
    const float addB = low ? 0.0f : 0.5f;

    float cst[8], hreg[8];
    #pragma unroll
    for (int k = 0; k < 8; ++k) { cst[k] = 0.0f; hreg[k] = 0.0f; }
    v2f bm0; bm0[0] = 0.0f; bm0[1] = 0.0f;   // B-matrix chunk0 (h == 0)
    v2f bm1; bm1[0] = 0.0f; bm1[1] = 0.0f;   // B-matrix chunk1

    const float* sp = seq + (size_t)s * SEQ_T;

    // software pipeline: keep one float4 (4 timesteps) in flight
    float4 xq = *(const float4*)(sp);

    for (int t = 0; t < SEQ_T; t += 4) {
        const float xs[4] = {xq.x, xq.y, xq.z, xq.w};
        // issue next quad's load now; its latency hides behind 4 dependent steps
        const int tn = (t + 4 < SEQ_T) ? (t + 4) : t;
        xq = *(const float4*)(sp + tn);
        // warm the next cacheline (64 steps ahead) toward the WGP
        __builtin_prefetch(sp + ((t + 64 < SEQ_T) ? (t + 64) : t), 0, 3);

        #pragma unroll
        for (int u = 0; u < 4; ++u) {
            const float x = xs[u];
            // C accumulator = x * W_ih + (b_ih + b_hh), already in D-layout
            v8f c0, c1;
            #pragma unroll
            for (int k = 0; k < 8; ++k) {
                c0[k] = fmaf(x, xw0[k], xb0[k]);
                c1[k] = fmaf(x, xw1[k], xb1[k]);
            }
            // D(32x16) = W_hh(32x8) x h^T(8x16) + C : K=8 via two chained K=4 WMMAs
            v8f d0 = __builtin_amdgcn_wmma_f32_16x16x4_f32(false, a00, false, bm0, (short)0, c0, false, false);
            d0     = __builtin_amdgcn_wmma_f32_16x16x4_f32(false, a01, false, bm1, (short)0, d0, false, false);
            v8f d1 = __builtin_amdgcn_wmma_f32_16x16x4_f32(false, a10, false, bm0, (short)0, c1, false, false);
            d1     = __builtin_amdgcn_wmma_f32_16x16x4_f32(false, a11, false, bm1, (short)0, d1, false, false);

            // activations: tile0 = sigmoid everywhere (i|f); tile1 = tanh|sigmoid (g|o)
            float act0[8], act1[8], X[8], Y[8];
            #pragma unroll
            for (int k = 0; k < 8; ++k) {
                act0[k] = fmaf(0.5f, fast_tanhf(0.5f * d0[k]), 0.5f);
                act1[k] = fmaf(sclA, fast_tanhf(sclA * d1[k]), addB);
            }
            #pragma unroll
            for (int k = 0; k < 8; ++k) {
                X[k] = swapx16(act0[k]);   // low: f, high: i
                Y[k] = swapx16(act1[k]);   // low: o, high: g
            }
            // symmetric elementwise update -> c,h replicated in both halves
            #pragma unroll
            for (int k = 0; k < 8; ++k) {
                const float ig = low ? (act0[k] * act1[k]) : (X[k] * Y[k]); // i*g
                const float fv = low ? X[k] : act0[k];                      // f
                const float ov = low ? Y[k] : act1[k];                      // o
                cst[k]  = fmaf(fv, cst[k], ig);
                hreg[k] = ov * fast_tanhf(cst[k]);
            }
            // rebuild B (h^T in B-layout): lane L, vgpr v -> h[L%16][4c + 2*(L/16) + v]
            bm0[0] = low ? hreg[0] : hreg[2];
            bm0[1] = low ? hreg[1] : hreg[3];
            bm1[0] = low ? hreg[4] : hreg[6];
            bm1[1] = low ? hreg[5] : hreg[7];
        }
    }

    // ---- head: tanh(h W1^T + b1) W2^T + b2 ; softplus on theta/sigma ----
    float z[8];
    #pragma unroll
    for (int j = 0; j < 8; ++j) {
        float acc = b1p[j];
        #pragma unroll
        for (int k = 0; k < 8; ++k) acc = fmaf(W1p[j * 8 + k], hreg[k], acc);
        z[j] = fast_tanhf(acc);
    }
    float raw[3];
    #pragma unroll
    for (int r = 0; r < 3; ++r) {
        float acc = b2p[r];
        #pragma unroll
        for (int j = 0; j < 8; ++j) acc = fmaf(W2p[r * 8 + j], z[j], acc);
        raw[r] = acc;
    }
    const float theta = softplusf(raw[0]) + 1e-4f;
    const float mu    = raw[1];
    const float sigma = softplusf(raw[2]) + 1e-4f;

    if (low && (waveG * 16 + l16) < Btotal) {
        out[(size_t)s * 3 + 0] = theta;
        out[(size_t)s * 3 + 1] = mu;
        out[(size_t)s * 3 + 2] = sigma;
    }
}

extern "C" void kernel_launch(void* const* d_in, const int* in_sizes, int n_in,
                              void* d_out, int out_size, void* d_ws, size_t ws_size,
                              hipStream_t stream) {
    const float* seq  = (const float*)d_in[0];
    const float* W_ih = (const float*)d_in[1];
    const float* W_hh = (const float*)d_in[2];
    const float* b_ih = (const float*)d_in[3];
    const float* b_hh = (const float*)d_in[4];
    const float* W1   = (const float*)d_in[5];
    const float* b1   = (const float*)d_in[6];
    const float* W2   = (const float*)d_in[7];
    const float* b2   = (const float*)d_in[8];
    float* out = (float*)d_out;

    const int B = in_sizes[0] / SEQ_T;           // 16384
    const int samplesPerBlock = 32;              // 2 waves * 16 samples
    const int blocks = (B + samplesPerBlock - 1) / samplesPerBlock;  // 512 workgroups

    NeuralOUEstimator_lstm_wmma_kernel<<<blocks, 64, 0, stream>>>(
        seq, W_ih, W_hh, b_ih, b_hh, W1, b1, W2, b2, out, B);
}